// SparseLayer_81801947119923
// MI455X (gfx1250) — compile-verified
//
#include <hip/hip_runtime.h>
#include <math.h>

// ---------------------------------------------------------------------------
// Sparse coding layer (LCA + AdamW) for MI455X / gfx1250.
// GEMMs on V_WMMA_F32_16X16X4_F32 (native fp32 matrix path — matches the fp32
// reference numerics). Double-buffered LDS ping-pong hides global latency;
// fragment software-pipelining hides ds_load latency behind the WMMA pipe.
// Iterative state (u/m/v, act, G) is workspace-resident (~88 MB < 192 MB L2).
// ---------------------------------------------------------------------------

typedef __attribute__((ext_vector_type(2))) float v2f;
typedef __attribute__((ext_vector_type(8))) float v8f;

#define BM 128
#define BN 128
#define BK 32
#define LDSA 36   // padded LDS row stride (floats); stride 36 mod 64 banks -> conflict-free

// ---- global -> register staging (float4 x4 = 16 floats per thread) ----
__device__ __forceinline__ void ldg4(const float* __restrict__ gp, float4 r[4]) {
  const float4* g4 = reinterpret_cast<const float4*>(gp);
  r[0] = g4[0]; r[1] = g4[1]; r[2] = g4[2]; r[3] = g4[3];
}

__device__ __forceinline__ void unpack16(const float4 r[4], float* t) {
  *reinterpret_cast<float4*>(t + 0)  = r[0];
  *reinterpret_cast<float4*>(t + 4)  = r[1];
  *reinterpret_cast<float4*>(t + 8)  = r[2];
  *reinterpret_cast<float4*>(t + 12) = r[3];
}

// ---- per-thread global tile addressing ----
template<int TA>
__device__ __forceinline__ const float* a_gptr(const float* __restrict__ A, int lda,
                                               int tileM, int k0, int tid) {
  if (TA == 0) {                       // A[m,k] row-major
    const int row = tid >> 1, kc = (tid & 1) << 4;
    return A + (size_t)(tileM + row) * lda + (k0 + kc);
  } else {                             // A[m,k] = Aglob[k*lda + m]
    const int kk = tid >> 3, mm = (tid & 7) << 4;
    return A + (size_t)(k0 + kk) * lda + (tileM + mm);
  }
}

template<int TB>
__device__ __forceinline__ const float* b_gptr(const float* __restrict__ B, int ldb,
                                               int tileN, int k0, int tid) {
  if (TB == 0) {                       // B[k,n] row-major
    const int kk = tid >> 3, nn = (tid & 7) << 4;
    return B + (size_t)(k0 + kk) * ldb + (tileN + nn);
  } else {                             // B[k,n] = Bglob[n*ldb + k]
    const int row = tid >> 1, kc = (tid & 1) << 4;
    return B + (size_t)(tileN + row) * ldb + (k0 + kc);
  }
}

// ---- register -> LDS stores (As: m-major [m][k]; Bs: n-major [n][k]) ----
template<int TA>
__device__ __forceinline__ void a_store(float* __restrict__ As, const float4 r[4], int tid) {
  alignas(16) float t[16];
  unpack16(r, t);
  if (TA == 0) {
    const int row = tid >> 1, kc = (tid & 1) << 4;
    float* s = &As[row * LDSA + kc];
    #pragma unroll
    for (int j = 0; j < 16; ++j) s[j] = t[j];
  } else {
    const int kk = tid >> 3, mm = (tid & 7) << 4;
    #pragma unroll
    for (int j = 0; j < 16; ++j) As[(mm + j) * LDSA + kk] = t[j];
  }
}

template<int TB>
__device__ __forceinline__ void b_store(float* __restrict__ Bs, const float4 r[4], int tid) {
  alignas(16) float t[16];
  unpack16(r, t);
  if (TB == 0) {
    const int kk = tid >> 3, nn = (tid & 7) << 4;
    #pragma unroll
    for (int j = 0; j < 16; ++j) Bs[(nn + j) * LDSA + kk] = t[j];
  } else {
    const int row = tid >> 1, kc = (tid & 1) << 4;
    float* s = &Bs[row * LDSA + kc];
    #pragma unroll
    for (int j = 0; j < 16; ++j) s[j] = t[j];
  }
}

// ---- LDS -> WMMA fragments (16x4 fp32: lane = M/N idx, VGPR pair = K 2*(lane/16)+{0,1})
__device__ __forceinline__ void load_frags(const float* __restrict__ as,
                                           const float* __restrict__ bs,
                                           int kf, int wm, int wn, int lrow,
                                           v2f af[4], v2f bf[2]) {
  #pragma unroll
  for (int mi = 0; mi < 4; ++mi) {
    const float* s = &as[(wm * 64 + mi * 16 + lrow) * LDSA + kf];
    af[mi].x = s[0]; af[mi].y = s[1];
  }
  #pragma unroll
  for (int ni = 0; ni < 2; ++ni) {
    const float* s = &bs[(wn * 32 + ni * 16 + lrow) * LDSA + kf];
    bf[ni].x = s[0]; bf[ni].y = s[1];
  }
}

// EPI: 0 = C[idx] = acc
//      1 = fused AdamW update of (u,m,v) using g = u - excite + acc
//      2 = C[idx] = acc - I   (for G = PhiT*Phi - I)
template<int TA, int TB, int EPI>
__global__ __launch_bounds__(256)
void wmma_gemm_f32(const float* __restrict__ A, const float* __restrict__ B,
                   float* __restrict__ C, int M, int N, int K,
                   int lda, int ldb, int ldc,
                   const float* __restrict__ excite,
                   float* __restrict__ ubuf, float* __restrict__ mbuf,
                   float* __restrict__ vbuf, float rc1, float rc2) {
  (void)M; (void)N;
  __shared__ float As[2][BM * LDSA];
  __shared__ float Bs[2][BN * LDSA];

  const int tid  = threadIdx.x;
  const int lane = tid & 31;
  const int wave = tid >> 5;        // 0..7
  const int half = lane >> 4;       // 0..1
  const int lrow = lane & 15;       // 0..15
  const int wm   = wave >> 2;       // 0..1 -> 64-row slab
  const int wn   = wave & 3;        // 0..3 -> 32-col slab
  const int tileM = blockIdx.y * BM;
  const int tileN = blockIdx.x * BN;

  const v8f vzero = {0.f, 0.f, 0.f, 0.f, 0.f, 0.f, 0.f, 0.f};
  v8f acc[4][2];
  #pragma unroll
  for (int mi = 0; mi < 4; ++mi)
    #pragma unroll
    for (int ni = 0; ni < 2; ++ni)
      acc[mi][ni] = vzero;

  float4 ra[4], rb[4];

  // prologue: stage chunk 0
  ldg4(a_gptr<TA>(A, lda, tileM, 0, tid), ra);
  ldg4(b_gptr<TB>(B, ldb, tileN, 0, tid), rb);
  a_store<TA>(&As[0][0], ra, tid);
  b_store<TB>(&Bs[0][0], rb, tid);
  __syncthreads();

  for (int k0 = 0; k0 < K; k0 += BK) {
    const int  cur  = (k0 / BK) & 1;
    const bool more = (k0 + BK) < K;

    // issue next chunk's global loads early; waits sink to the LDS store below
    if (more) {
      ldg4(a_gptr<TA>(A, lda, tileM, k0 + BK, tid), ra);
      ldg4(b_gptr<TB>(B, ldb, tileN, k0 + BK, tid), rb);
    }

    const float* __restrict__ as = &As[cur][0];
    const float* __restrict__ bs = &Bs[cur][0];

    // fragment-pipelined compute: preload k-step kk+4 while issuing WMMAs for kk
    v2f afc[4], bfc[2], afn[4], bfn[2];
    load_frags(as, bs, (half << 1), wm, wn, lrow, afc, bfc);
    #pragma unroll
    for (int kk = 0; kk < BK; kk += 4) {
      if (kk + 4 < BK)
        load_frags(as, bs, kk + 4 + (half << 1), wm, wn, lrow, afn, bfn);
      #pragma unroll
      for (int mi = 0; mi < 4; ++mi)
        #pragma unroll
        for (int ni = 0; ni < 2; ++ni)
          acc[mi][ni] = __builtin_amdgcn_wmma_f32_16x16x4_f32(
              false, afc[mi], false, bfc[ni], (short)0, acc[mi][ni], false, false);
      #pragma unroll
      for (int mi = 0; mi < 4; ++mi) afc[mi] = afn[mi];
      #pragma unroll
      for (int ni = 0; ni < 2; ++ni) bfc[ni] = bfn[ni];
    }

    // store next chunk into the other buffer (read of that buffer finished at
    // the barrier ending the previous iteration); one barrier per chunk
    if (more) {
      a_store<TA>(&As[1 - cur][0], ra, tid);
      b_store<TB>(&Bs[1 - cur][0], rb, tid);
      __syncthreads();
    }
  }

  // ---- epilogue: D layout: lane -> N = lane%16, VGPR r -> M = r + 8*(lane/16)
  #pragma unroll
  for (int mi = 0; mi < 4; ++mi) {
    #pragma unroll
    for (int ni = 0; ni < 2; ++ni) {
      const int gcol = tileN + wn * 32 + ni * 16 + lrow;
      #pragma unroll
      for (int r = 0; r < 8; ++r) {
        const int grow = tileM + wm * 64 + mi * 16 + r + 8 * half;
        const size_t idx = (size_t)grow * ldc + gcol;
        const float val = acc[mi][ni][r];
        if (EPI == 0) {
          C[idx] = val;
        } else if (EPI == 2) {
          C[idx] = val - ((grow == gcol) ? 1.0f : 0.0f);
        } else {
          // AdamW: g = u - excite + (act@G);  u *= (1 - lr*wd); then Adam step
          const float uo = ubuf[idx];
          const float g  = uo - excite[idx] + val;
          const float mn = 0.9f * mbuf[idx] + 0.1f * g;
          const float vn = 0.999f * vbuf[idx] + 0.001f * (g * g);
          float un = uo * (1.0f - 0.1f * 1e-2f);
          un -= 0.1f * (mn * rc1) / (sqrtf(vn * rc2) + 1e-8f);
          ubuf[idx] = un; mbuf[idx] = mn; vbuf[idx] = vn;
        }
      }
    }
  }
}

__global__ void zero_kernel(float* __restrict__ p, int n4) {
  int i = blockIdx.x * blockDim.x + threadIdx.x;
  if (i < n4) {
    float4 z; z.x = 0.f; z.y = 0.f; z.z = 0.f; z.w = 0.f;
    reinterpret_cast<float4*>(p)[i] = z;
  }
}

__global__ void shrink_kernel(const float* __restrict__ u, float* __restrict__ act, int n4) {
  int i = blockIdx.x * blockDim.x + threadIdx.x;
  if (i < n4) {
    float4 x = reinterpret_cast<const float4*>(u)[i];
    float4 y;
    y.x = fmaxf(x.x - 0.25f, 0.f);
    y.y = fmaxf(x.y - 0.25f, 0.f);
    y.z = fmaxf(x.z - 0.25f, 0.f);
    y.w = fmaxf(x.w - 0.25f, 0.f);
    reinterpret_cast<float4*>(act)[i] = y;
  }
}

__global__ void shrink2_kernel(const float* __restrict__ u, float* __restrict__ act,
                               float* __restrict__ acts_out, int n4) {
  int i = blockIdx.x * blockDim.x + threadIdx.x;
  if (i < n4) {
    float4 x = reinterpret_cast<const float4*>(u)[i];
    float4 y;
    y.x = fmaxf(x.x - 0.25f, 0.f);
    y.y = fmaxf(x.y - 0.25f, 0.f);
    y.z = fmaxf(x.z - 0.25f, 0.f);
    y.w = fmaxf(x.w - 0.25f, 0.f);
    reinterpret_cast<float4*>(act)[i] = y;
    reinterpret_cast<float4*>(acts_out)[i] = y;
  }
}

extern "C" void kernel_launch(void* const* d_in, const int* in_sizes, int n_in,
                              void* d_out, int out_size, void* d_ws, size_t ws_size,
                              hipStream_t stream) {
  (void)in_sizes; (void)n_in; (void)out_size; (void)ws_size;
  const int Bsz = 4096, WH = 4096, F = 1024;
  const size_t NBF = (size_t)Bsz * F;                  // 4,194,304

  const float* images  = (const float*)d_in[0];        // [B, WH] row-major
  const float* filters = (const float*)d_in[1];        // Phi: [WH, F] row-major

  float* out      = (float*)d_out;
  float* recon    = out;                               // [B, WH]
  float* acts_out = out + (size_t)Bsz * WH;            // [B, F]

  float* ws     = (float*)d_ws;                        // ~88 MB total
  float* excite = ws;                                  // [B, F]
  float* u      = ws + 1 * NBF;                        // [B, F]
  float* mbuf   = ws + 2 * NBF;                        // [B, F]
  float* vbuf   = ws + 3 * NBF;                        // [B, F]
  float* act    = ws + 4 * NBF;                        // [B, F]
  float* G      = ws + 5 * NBF;                        // [F, F] = PhiT*Phi - I

  const dim3 blk(256);

  // u = m = v = 0 (contiguous 3*NBF floats)
  {
    int n4z = (int)(3 * NBF / 4);
    zero_kernel<<<(n4z + 255) / 256, blk, 0, stream>>>(u, n4z);
  }

  // G = PhiT @ Phi - I    (M=N=F, K=WH; A = Phi transposed)
  wmma_gemm_f32<1, 0, 2><<<dim3(F / BN, F / BM), blk, 0, stream>>>(
      filters, filters, G, F, F, WH, F, F, F,
      nullptr, nullptr, nullptr, nullptr, 0.f, 0.f);

  // excite = X @ Phi      (M=B, N=F, K=WH)
  wmma_gemm_f32<0, 0, 0><<<dim3(F / BN, Bsz / BM), blk, 0, stream>>>(
      images, filters, excite, Bsz, F, WH, WH, F, F,
      nullptr, nullptr, nullptr, nullptr, 0.f, 0.f);

  // 50 LCA + AdamW iterations
  const int n4 = (int)(NBF / 4);
  for (int t = 1; t <= 50; ++t) {
    shrink_kernel<<<(n4 + 255) / 256, blk, 0, stream>>>(u, act, n4);
    const float rc1 = 1.0f / (1.0f - powf(0.9f,   (float)t));
    const float rc2 = 1.0f / (1.0f - powf(0.999f, (float)t));
    // r = act @ G, fused with AdamW update of (u, m, v)
    wmma_gemm_f32<0, 0, 1><<<dim3(F / BN, Bsz / BM), blk, 0, stream>>>(
        act, G, u, Bsz, F, F, F, F, F,
        excite, u, mbuf, vbuf, rc1, rc2);
  }

  // final activations -> act (for recon GEMM) and d_out acts region
  shrink2_kernel<<<(n4 + 255) / 256, blk, 0, stream>>>(u, act, acts_out, n4);

  // recon = act @ PhiT    (M=B, N=WH, K=F; B[k,n] = Phi[n*F + k])
  wmma_gemm_f32<0, 1, 0><<<dim3(WH / BN, Bsz / BM), blk, 0, stream>>>(
      act, filters, recon, Bsz, WH, F, F, F, WH,
      nullptr, nullptr, nullptr, nullptr, 0.f, 0.f);
}